// Model_41558103556402
// MI455X (gfx1250) — compile-verified
//
#include <hip/hip_runtime.h>

typedef float v2f __attribute__((ext_vector_type(2)));
typedef float v8f __attribute__((ext_vector_type(8)));

#define NMAT 2048
#define BPAIR 32
#define DDIM 16
#define LSTR 17            // padded row stride (floats) to avoid LDS bank conflicts
#define MATF (DDIM*LSTR)
#define WAVES 4
#define NBUF 12
#define NIT_SQRT 12
#define NIT_INV 14
#define NCOEF 24

// Wave-level LDS sync: LDS ops from one wave complete in order (DScnt), so a
// dscnt-wait + compiler memory fence is enough for intra-wave cross-lane
// communication through the wave-private LDS region. No block barrier needed.
#define WSYNC() asm volatile("s_wait_dscnt 0" ::: "memory")

static __device__ __forceinline__ int laneid() { return (int)(threadIdx.x & 31u); }

// ---- WMMA fragment builders (CDNA5 f32 16x16x4 layouts, ISA 7.12.2) ----
// A 16x4 (f32): lanes 0-15 -> (K=k0,k0+1), lanes 16-31 -> (K=k0+2,k0+3), M=lane&15
template<bool TA>
static __device__ __forceinline__ v2f frag_a(const float* M, int k0, float s) {
  int l = laneid();
  int m  = l & 15;
  int kb = k0 + ((l >> 4) << 1);
  v2f a;
  if (!TA) { a.x = M[m*LSTR + kb];  a.y = M[m*LSTR + kb + 1]; }
  else     { a.x = M[kb*LSTR + m];  a.y = M[(kb+1)*LSTR + m]; }   // transposed read
  a.x *= s; a.y *= s;
  return a;
}

// B 4x16 (f32): rows striped across lanes; mirror of the A pattern
template<bool TB>
static __device__ __forceinline__ v2f frag_b(const float* M, int k0) {
  int l = laneid();
  int n  = l & 15;
  int kb = k0 + ((l >> 4) << 1);
  v2f b;
  if (!TB) { b.x = M[kb*LSTR + n];  b.y = M[(kb+1)*LSTR + n]; }
  else     { b.x = M[n*LSTR + kb];  b.y = M[n*LSTR + kb + 1]; }   // reads B^T
  return b;
}

// C += (s*A[opt T]) x B[opt T]  as four chained K=4 WMMAs
template<bool TA, bool TB>
static __device__ __forceinline__ v8f mm_acc(v8f c, const float* A, float sa, const float* B) {
#pragma unroll
  for (int k0 = 0; k0 < DDIM; k0 += 4) {
    v2f a = frag_a<TA>(A, k0, sa);
    v2f b = frag_b<TB>(B, k0);
    c = __builtin_amdgcn_wmma_f32_16x16x4_f32(false, a, false, b, (short)0, c, false, false);
  }
  return c;
}

// C/D 16x16 f32 layout: VGPR v, lanes 0-15 -> M=v, lanes 16-31 -> M=v+8, N=lane&15
static __device__ __forceinline__ void store_c(float* M, v8f c) {
  int l  = laneid();
  int n  = l & 15;
  int mb = (l >> 4) * 8;
#pragma unroll
  for (int v = 0; v < 8; ++v) M[(mb + v)*LSTR + n] = c[v];
}

// load a 16x16 matrix from LDS directly in C/D fragment layout (scaled):
// lets us fold "alpha*X + A*B" into the WMMA accumulator with zero extra passes.
static __device__ __forceinline__ v8f load_c(const float* M, float s) {
  int l  = laneid();
  int n  = l & 15;
  int mb = (l >> 4) * 8;
  v8f c;
#pragma unroll
  for (int v = 0; v < 8; ++v) c[v] = M[(mb + v)*LSTR + n] * s;
  return c;
}

// real C = A*B
template<bool TA, bool TB>
static __device__ __forceinline__ void rmatmul(float* C, const float* A, const float* B) {
  WSYNC();
  v8f c = {};
  c = mm_acc<TA,TB>(c, A, 1.f, B);
  store_c(C, c);
}

// real C = s0*C0 + sa*(A*B)  (accumulator-folded axpy, no elementwise pass)
static __device__ __forceinline__ void rmatmul_axpy(float* C, const float* C0, float s0,
                                                    const float* A, float sa, const float* B) {
  WSYNC();
  v8f c = load_c(C0, s0);
  c = mm_acc<false,false>(c, A, sa, B);
  store_c(C, c);
}

// complex C = (Ar + iAi)(Br + iBi) + diag*I (diag added in-register, on real part)
static __device__ __forceinline__ void cmatmul_diag(float* Cr, float* Ci,
    const float* Ar, const float* Ai, const float* Br, const float* Bi, float diag) {
  WSYNC();
  v8f cr = {}, ci = {};
  cr = mm_acc<false,false>(cr, Ar,  1.f, Br);
  cr = mm_acc<false,false>(cr, Ai, -1.f, Bi);
  ci = mm_acc<false,false>(ci, Ar,  1.f, Bi);
  ci = mm_acc<false,false>(ci, Ai,  1.f, Br);
  if (diag != 0.f) {
    int l = laneid(); int n = l & 15; int mb = (l >> 4) * 8;
#pragma unroll
    for (int v = 0; v < 8; ++v) if (mb + v == n) cr[v] += diag;
  }
  store_c(Cr, cr);
  store_c(Ci, ci);
}

// complex Newton step for inverse: D = 2V - V*P (all folded into accumulators)
static __device__ __forceinline__ void inv_newton_step(float* Dr, float* Di,
    const float* Vr, const float* Vi, const float* Pr, const float* Pi) {
  WSYNC();
  v8f cr = load_c(Vr, 2.f);
  cr = mm_acc<false,false>(cr, Vr, -1.f, Pr);
  cr = mm_acc<false,false>(cr, Vi,  1.f, Pi);
  v8f ci = load_c(Vi, 2.f);
  ci = mm_acc<false,false>(ci, Vr, -1.f, Pi);
  ci = mm_acc<false,false>(ci, Vi, -1.f, Pr);
  store_c(Dr, cr);
  store_c(Di, ci);
}

__global__ __launch_bounds__(WAVES*32) void siegel_loss_kernel(
    const float* __restrict__ embeds,   // [N][2][16][16] f32
    const float* __restrict__ gdist,    // [N][N] f32
    float* __restrict__ out,            // [1] f32 (pre-zeroed)
    int npairs) {
  __shared__ float lds[WAVES][NBUF][MATF];
  const int wave = (int)(threadIdx.x >> 5);
  const int lane = laneid();
  float (*b)[MATF] = lds[wave];

  // ---- reconstruct (src,dst) analytically from pair id ----
  int p = (int)blockIdx.x * WAVES + wave;
  if (p >= npairs) p = npairs - 1;
  int rem = p, ia = 0;
#pragma unroll 1
  for (; ia < BPAIR - 1; ++ia) {
    int cnt = NMAT - 1 - ia;
    if (rem < cnt) break;
    rem -= cnt;
  }
  int ib = ia + 1 + rem;

  const float* X1 = embeds + ((size_t)ia * 2 + 0) * 256;
  const float* Y1 = embeds + ((size_t)ia * 2 + 1) * 256;
  const float* X2 = embeds + ((size_t)ib * 2 + 0) * 256;
  const float* Y2 = embeds + ((size_t)ib * 2 + 1) * 256;
  __builtin_prefetch(X1 + lane * 8, 0, 3);     // global_prefetch_b8
  __builtin_prefetch(X2 + lane * 8, 0, 3);
  const float gd  = gdist[(size_t)ia * NMAT + ib];

  // ---- phase 1: S = Y2^(-1/2) via coupled Newton-Schulz (accumulator-folded) ----
  // ping-pong: y in {b0,b1}, z in {b2,b3}, u = b4
  float trc = 0.f;
#pragma unroll
  for (int i = 0; i < 8; ++i) {
    int e = lane + 32*i, m = e >> 4, n = e & 15;
    float y = Y2[e];
    if (m == n) trc += y;
  }
#pragma unroll
  for (int s = 16; s >= 1; s >>= 1) trc += __shfl_xor(trc, s, 32);
  const float invc = 1.f / trc;
#pragma unroll
  for (int i = 0; i < 8; ++i) {
    int e = lane + 32*i, m = e >> 4, n = e & 15;
    b[0][m*LSTR+n] = Y2[e] * invc;              // Y0 = Y2/tr (L2-cached reload)
    b[2][m*LSTR+n] = (m == n) ? 1.f : 0.f;      // Z0 = I
  }

  float *y = b[0], *yn = b[1], *z = b[2], *zn = b[3], *u = b[4];
#pragma unroll 1
  for (int it = 0; it < NIT_SQRT; ++it) {
    rmatmul<false,false>(u, z, y);              // U = Z*Y
    rmatmul_axpy(yn, y, 1.5f, y, -0.5f, u);     // Ynew = 1.5Y - 0.5*Y*U
    rmatmul_axpy(zn, z, 1.5f, u, -0.5f, z);     // Znew = 1.5Z - 0.5*U*Z
    float* t;
    t = y; y = yn; yn = t;
    t = z; z = zn; zn = t;
  }
  // S = Z / sqrt(trc)  -> b5
  WSYNC();
  {
    float rs = rsqrtf(trc);
#pragma unroll
    for (int i = 0; i < 8; ++i) {
      int e = lane + 32*i, m = e >> 4, n = e & 15;
      b[5][m*LSTR+n] = z[m*LSTR+n] * rs;
    }
  }

  // ---- phase 2: Z3 = S*(X1-X2 + iY1)*S ----
  WSYNC();
#pragma unroll
  for (int i = 0; i < 8; ++i) {
    int e = lane + 32*i, m = e >> 4, n = e & 15;
    b[0][m*LSTR+n] = X1[e] - X2[e];   // Dr
    b[1][m*LSTR+n] = Y1[e];           // Di
  }
  rmatmul<false,false>(b[2], b[5], b[0]);   // S*Dr
  rmatmul<false,false>(b[6], b[2], b[5]);   // Zr = (S*Dr)*S
  rmatmul<false,false>(b[2], b[5], b[1]);   // S*Di
  rmatmul<false,false>(b[7], b[2], b[5]);   // Zi = (S*Di)*S

  // G = Z3 + iI (Gr=b6, Gi=b8), H = Z3 - iI (Hr=b6, Hi=b9)
  WSYNC();
#pragma unroll
  for (int i = 0; i < 8; ++i) {
    int e = lane + 32*i, m = e >> 4, n = e & 15;
    float zi = b[7][m*LSTR+n];
    float d  = (m==n) ? 1.f : 0.f;
    b[8][m*LSTR+n] = zi + d;
    b[9][m*LSTR+n] = zi - d;
  }

  // ---- phase 3: V = G^{-1} via Newton  V <- 2V - V*(G*V),  V0 = G^H/||G||_F^2 ----
  WSYNC();
  float al = 0.f;
#pragma unroll
  for (int i = 0; i < 8; ++i) {
    int e = lane + 32*i, m = e >> 4, n = e & 15;
    float gr = b[6][m*LSTR+n], gi = b[8][m*LSTR+n];
    al += gr*gr + gi*gi;
  }
#pragma unroll
  for (int s = 16; s >= 1; s >>= 1) al += __shfl_xor(al, s, 32);
  {
    float ina = 1.f / al;
#pragma unroll
    for (int i = 0; i < 8; ++i) {               // V0 = G^H/al  (Vr=b10, Vi=b11)
      int e = lane + 32*i, m = e >> 4, n = e & 15;
      b[10][m*LSTR+n] =  b[6][n*LSTR+m] * ina;
      b[11][m*LSTR+n] = -b[8][n*LSTR+m] * ina;
    }
  }
  float *vr = b[10], *vi = b[11], *wr = b[0], *wi = b[1];
#pragma unroll 1
  for (int it = 0; it < NIT_INV; ++it) {
    cmatmul_diag(b[2], b[3], b[6], b[8], vr, vi, 0.f);   // P = G*V
    inv_newton_step(wr, wi, vr, vi, b[2], b[3]);         // Vn = 2V - V*P
    float* t;
    t = vr; vr = wr; wr = t;
    t = vi; vi = wi; wi = t;
  }

  // K = V*H (Kr=b2, Ki=b3);  W = K^T, M = W^H W = conj(K)*K^T (Hermitian, eig = s^2)
  cmatmul_diag(b[2], b[3], vr, vi, b[6], b[9], 0.f);
  WSYNC();
  {
    v8f mr = {}, mi = {};
    mr = mm_acc<false,true>(mr, b[2],  1.f, b[2]);    // Kr*Kr^T
    mr = mm_acc<false,true>(mr, b[3],  1.f, b[3]);    // + Ki*Ki^T
    mi = mm_acc<false,true>(mi, b[2],  1.f, b[3]);    // Kr*Ki^T
    mi = mm_acc<false,true>(mi, b[3], -1.f, b[2]);    // - Ki*Kr^T
    const float clip = 0.998f;                        // spectral clip s^2 < 1 (cf. EPS)
#pragma unroll
    for (int v = 0; v < 8; ++v) { mr[v] *= clip; mi[v] *= clip; }
    store_c(b[4], mr);                                // Mr
    store_c(b[5], mi);                                // Mi
  }

  // ---- phase 4: dist^2 = 4*tr(sum_n c_n M^n), c_n = (1/n) sum_{k<n} 1/(2k+1) ----
  float h = 0.f;
#pragma unroll 1
  for (int n2 = 1; n2 <= NCOEF; ++n2) h += 1.f / (float)(2*n2 - 1);
  float cn = h / (float)NCOEF;
  WSYNC();
#pragma unroll
  for (int i = 0; i < 8; ++i) {                       // P = c_N * I  (Pr=b10, Pi=b11)
    int e = lane + 32*i, m = e >> 4, n = e & 15;
    b[10][m*LSTR+n] = (m==n) ? cn : 0.f;
    b[11][m*LSTR+n] = 0.f;
  }
  float *pr = b[10], *pi = b[11], *qr = b[0], *qi = b[1];
#pragma unroll 1
  for (int n2 = NCOEF - 1; n2 >= 1; --n2) {           // Horner: P <- M*P + c_n I
    h -= 1.f / (float)(2*n2 + 1);
    cn = h / (float)n2;
    cmatmul_diag(qr, qi, b[4], b[5], pr, pi, cn);     // +c_n I folded in-register
    float* t;
    t = pr; pr = qr; qr = t;
    t = pi; pi = qi; qi = t;
  }
  // tr(M*P) without the final matmul: sum_ij Mr_ij Pr_ji - Mi_ij Pi_ji
  WSYNC();
  float tr = 0.f;
#pragma unroll
  for (int i = 0; i < 8; ++i) {
    int e = lane + 32*i, m = e >> 4, n = e & 15;
    tr += b[4][m*LSTR+n] * pr[n*LSTR+m] - b[5][m*LSTR+n] * pi[n*LSTR+m];
  }
#pragma unroll
  for (int s = 16; s >= 1; s >>= 1) tr += __shfl_xor(tr, s, 32);

  if (lane == 0) {
    float dist2 = 4.f * tr;                           // = sum v_i^2 (md^2)
    float loss  = fabsf(dist2 / (gd * gd) - 1.f);     // |(md/gd)^2 - 1|
    atomicAdd(out, loss);
  }
}

extern "C" void kernel_launch(void* const* d_in, const int* in_sizes, int n_in,
                              void* d_out, int out_size, void* d_ws, size_t ws_size,
                              hipStream_t stream) {
  (void)in_sizes; (void)n_in; (void)d_ws; (void)ws_size;
  const float* embeds = (const float*)d_in[0];
  const float* gdist  = (const float*)d_in[1];
  float* out = (float*)d_out;

  int npairs = 0;
  for (int a = 0; a < BPAIR; ++a) npairs += NMAT - 1 - a;   // 65008 = 4 * 16252

  hipMemsetAsync(out, 0, sizeof(float) * (out_size > 0 ? out_size : 1), stream);
  int blocks = (npairs + WAVES - 1) / WAVES;
  siegel_loss_kernel<<<blocks, WAVES * 32, 0, stream>>>(embeds, gdist, out, npairs);
}